// Net_49177375539428
// MI455X (gfx1250) — compile-verified
//
#include <hip/hip_runtime.h>

typedef __attribute__((ext_vector_type(2))) float v2f;
typedef __attribute__((ext_vector_type(8))) float v8f;

#define DIM 128
#define NN  128
#define EC  128
#define GC  8
#define VD  300

// ---------------------------------------------------------------------------
// Kernel 1: base[n][d] = data_vecs[data[n]] (300) @ data_weights (300x128) + b
// ---------------------------------------------------------------------------
__global__ __launch_bounds__(128) void k_base(const int* __restrict__ data,
                                              const float* __restrict__ dvecs,
                                              const float* __restrict__ dw,
                                              const float* __restrict__ db,
                                              float* __restrict__ base) {
  __shared__ float row[VD];
  const int n = blockIdx.x;
  const int tok = data[n];
  const float* src = dvecs + (size_t)tok * VD;
  for (int k = threadIdx.x; k < VD; k += 128) row[k] = src[k];
  __syncthreads();
  const int d = threadIdx.x;
  float s = db[d];
  for (int k = 0; k < VD; ++k) s += row[k] * dw[k * DIM + d];
  base[n * DIM + d] = s;
}

// ---------------------------------------------------------------------------
// 128x128x128 fp32 GEMM with V_WMMA_F32_16X16X4_F32.
// 8 waves: wave w owns rows [16w,16w+16); each wave does 8 col tiles, 32 K-steps.
// C[m][n] = sum_k A[m][k]*B[k][n] + bias[n]
// ---------------------------------------------------------------------------
__device__ inline void gemm128_wmma(const float* __restrict__ A,
                                    const float* __restrict__ B,
                                    const float* __restrict__ bias,
                                    float* __restrict__ C, int tid) {
  const int wave = tid >> 5;
  const int lane = tid & 31;
  const int ll   = lane & 15;
  const int lhi  = lane >> 4;          // 0: K=k0..k0+1, 1: K=k0+2..k0+3
  const int m0   = wave * 16;
  for (int nt = 0; nt < 8; ++nt) {
    const int n = nt * 16 + ll;
    v8f acc = {};
    const float* arow = A + (m0 + ll) * DIM + lhi * 2;   // A fragment rows
    const float* bcol = B + (lhi * 2) * DIM + n;         // B fragment cols
    #pragma unroll 4
    for (int k0 = 0; k0 < DIM; k0 += 4) {
      v2f a, b;
      a.x = arow[0];   a.y = arow[1];
      b.x = bcol[0];   b.y = bcol[DIM];
      acc = __builtin_amdgcn_wmma_f32_16x16x4_f32(false, a, false, b,
                                                  (short)0, acc, false, false);
      arow += 4;
      bcol += 4 * DIM;
    }
    const float bb = bias ? bias[n] : 0.0f;
    #pragma unroll
    for (int r = 0; r < 8; ++r)
      C[(m0 + r + lhi * 8) * DIM + n] = acc[r] + bb;
  }
}

// ---------------------------------------------------------------------------
// Kernel 2: one 256-thread block per graph; ascending-index tree fold.
// Rank-1 nodes: D-vector state + matvec. Pos->root path: (E,D) matrix + WMMA.
// ---------------------------------------------------------------------------
__global__ __launch_bounds__(256) void k_tree(const int* __restrict__ graphs,
                                              const int* __restrict__ edges,
                                              const int* __restrict__ posp,
                                              const float* __restrict__ ew,
                                              const float* __restrict__ eb,
                                              const float* __restrict__ sew,
                                              const float* __restrict__ sdw,
                                              const float* __restrict__ sbias,
                                              const float* __restrict__ base,
                                              float* __restrict__ accv_all,
                                              float* __restrict__ mat_all,
                                              float* __restrict__ out) {
  __shared__ int   s_parent[NN];
  __shared__ int   s_cc[NN];
  __shared__ int   s_eff[NN];
  __shared__ float s_ev[DIM];
  __shared__ float s_red[256];
  __shared__ float s_ds;

  const int j   = blockIdx.x;
  const int tid = threadIdx.x;
  const int pos = posp[0];

  float* accv  = accv_all + (size_t)j * (NN * DIM);
  float* mbuf0 = mat_all  + (size_t)j * (2 * EC * DIM);
  float* mbuf1 = mbuf0 + EC * DIM;

  if (tid < NN) s_cc[tid] = 0;
  if (tid == 0) {
    // Build parents with root chaining: each zero-offset root points at the
    // next-higher root; the last root gets -1 (the true root).
    int next_root = -1;
    for (int i = NN - 1; i >= 0; --i) {
      const int off = graphs[j * NN + i];
      if (off == 0) { s_parent[i] = next_root; next_root = i; }
      else           s_parent[i] = i + off;
    }
  }
  for (int t = tid; t < NN * DIM; t += 256) accv[t] = 0.0f;
  __syncthreads();
  if (tid < NN) {
    const int p = s_parent[tid];
    if (p >= 0 && p < NN) atomicAdd(&s_cc[p], 1);
    s_eff[tid] = (tid == pos) ? -1 : edges[tid];
  }
  __syncthreads();

  int    matOwner = -1;       // uniform across block
  float* Min  = mbuf0;
  float* Mout = mbuf1;
  float* res  = nullptr;

  for (int i = 0; i < NN; ++i) {
    const int  p     = s_parent[i];
    const int  eid   = s_eff[i];
    const bool isMat = (i == matOwner);
    const bool hasCh = (s_cc[i] > 0);

    if (isMat) {
      // finalize matrix state: Min += broadcast(base[i] + accv[i]); relu
      for (int t = tid; t < EC * DIM; t += 256) {
        const int d = t & (DIM - 1);
        float v = Min[t] + base[i * DIM + d] + accv[i * DIM + d];
        if (hasCh) v = fmaxf(v, 0.0f);
        Min[t] = v;
      }
      __syncthreads();
      if (eid >= 0) {
        gemm128_wmma(Min, ew + (size_t)eid * DIM * DIM,
                     eb + (size_t)eid * DIM, Mout, tid);
        float* t2 = Min; Min = Mout; Mout = t2;   // result now in Min
        if (p >= 0) matOwner = p;
        else        res = Min;                    // root transform result
      }
    } else {
      // finalize vector state
      if (tid < DIM) {
        float v = base[i * DIM + tid] + accv[i * DIM + tid];
        if (hasCh) v = fmaxf(v, 0.0f);
        s_ev[tid] = v;
      }
      __syncthreads();
      if (eid >= 0) {
        // matvec: v' = ev @ W[eid] + eb[eid], split-K over two thread halves
        {
          const int d = tid & (DIM - 1);
          const int h = tid >> 7;
          const float* Wp = ew + (size_t)eid * DIM * DIM + d;
          float s = 0.0f;
          const int k0 = h * 64;
          #pragma unroll 8
          for (int k = k0; k < k0 + 64; ++k) s += s_ev[k] * Wp[(size_t)k * DIM];
          s_red[tid] = s;
        }
        __syncthreads();
        if (tid < DIM) {
          const float v = s_red[tid] + s_red[tid + DIM] + eb[(size_t)eid * DIM + tid];
          if (p >= 0) accv[p * DIM + tid] += v;
          else
            for (int e = 0; e < EC; ++e) Mout[e * DIM + tid] = v;  // degenerate root
        }
        if (p < 0) res = Mout;
      } else {
        // pos node: per-edge-type batched transform -> full (E,D) matrix
        float* dst = (p >= 0) ? Min : Mout;
        for (int e = 0; e < EC; ++e) {
          const int d = tid & (DIM - 1);
          const int h = tid >> 7;
          const float* Wp = ew + (size_t)e * DIM * DIM + d;
          float s = 0.0f;
          const int k0 = h * 64;
          #pragma unroll 8
          for (int k = k0; k < k0 + 64; ++k) s += s_ev[k] * Wp[(size_t)k * DIM];
          s_red[tid] = s;
          __syncthreads();
          if (tid < DIM)
            dst[e * DIM + tid] = s_red[tid] + s_red[tid + DIM] + eb[(size_t)e * DIM + tid];
          __syncthreads();
        }
        if (p >= 0) matOwner = p;
        else        res = Mout;
      }
    }
    __syncthreads();
  }

  // d_score = score_bias + base[pos] . score_data_weights
  if (tid < DIM) s_red[tid] = base[pos * DIM + tid] * sdw[tid];
  __syncthreads();
  if (tid == 0) {
    float s = sbias[0];
    for (int d = 0; d < DIM; ++d) s += s_red[d];
    s_ds = s;
  }
  __syncthreads();

  if (tid < EC) {
    float s = s_ds;
    if (res) {
      const float* r = res + tid * DIM;
      for (int d = 0; d < DIM; ++d) s += r[d] * sew[d];
    }
    out[j * EC + tid] = s;
  }
}

// ---------------------------------------------------------------------------
extern "C" void kernel_launch(void* const* d_in, const int* in_sizes, int n_in,
                              void* d_out, int out_size, void* d_ws, size_t ws_size,
                              hipStream_t stream) {
  const int*   data  = (const int*)d_in[0];
  // d_in[1] = types (unused: single data_type in reference)
  const int*   graphs = (const int*)d_in[2];
  const int*   edges  = (const int*)d_in[3];
  const int*   posp   = (const int*)d_in[4];
  const float* dvecs  = (const float*)d_in[5];
  const float* dw     = (const float*)d_in[6];
  const float* db     = (const float*)d_in[7];
  const float* ew     = (const float*)d_in[8];
  const float* eb     = (const float*)d_in[9];
  const float* sew    = (const float*)d_in[10];
  const float* sdw    = (const float*)d_in[11];
  const float* sb     = (const float*)d_in[12];
  float* out = (float*)d_out;

  float* ws   = (float*)d_ws;
  float* base = ws;                         // 128*128
  float* accv = ws + NN * DIM;              // 8 * 128*128
  float* mats = accv + GC * NN * DIM;       // 8 * 2 * 128*128

  k_base<<<NN, 128, 0, stream>>>(data, dvecs, dw, db, base);
  k_tree<<<GC, 256, 0, stream>>>(graphs, edges, posp, ew, eb, sew, sdw, sb,
                                 base, accv, mats, out);
}